// Top2Gating_59605556134257
// MI455X (gfx1250) — compile-verified
//
#include <hip/hip_runtime.h>

#define TOKENS 8192
#define DIM    1024
#define NGATES 64
#define CAP    160
#define DISPATCH_ELEMS (TOKENS * NGATES * CAP)   // 83,886,080 floats per tensor

typedef __attribute__((ext_vector_type(2))) float v2f;
typedef __attribute__((ext_vector_type(4))) float v4f;
typedef __attribute__((ext_vector_type(8))) float v8f;

// ---------------------------------------------------------------------------
// Kernel 1: stream zeros over dispatch+combine (0.67 GB -> the real cost).
// b128 non-temporal stores: output far exceeds 192MB L2, so bypass-friendly.
// ---------------------------------------------------------------------------
__global__ void __launch_bounds__(256) zero_kernel(v4f* __restrict__ out4, long n4,
                                                   float* __restrict__ tail, int ntail) {
    long i      = (long)blockIdx.x * blockDim.x + threadIdx.x;
    long stride = (long)gridDim.x * blockDim.x;
    v4f z = {0.f, 0.f, 0.f, 0.f};
    for (; i < n4; i += stride)
        __builtin_nontemporal_store(z, &out4[i]);
    int t = blockIdx.x * blockDim.x + threadIdx.x;
    if (t < ntail) tail[t] = 0.f;
}

// ---------------------------------------------------------------------------
// Kernel 2: logits = x @ w_gating via V_WMMA_F32_16X16X4_F32 (exact fp32).
// One wave per 16x16 output tile; K loop of 256 WMMAs.
// A frag (16x4 f32): lane l<16 holds M=l, K=k..k+1; lane l>=16 holds K=k+2..k+3.
// B frag (4x16 f32): lane half selects K pair, lane&15 selects N.
// ---------------------------------------------------------------------------
__global__ void __launch_bounds__(256) gemm_wmma(const float* __restrict__ x,
                                                 const float* __restrict__ w,
                                                 float* __restrict__ logits) {
    int wave = blockIdx.x * (blockDim.x >> 5) + (threadIdx.x >> 5);
    int lane = threadIdx.x & 31;
    int tm   = (wave >> 2) << 4;          // 512 row tiles
    int tn   = (wave &  3) << 4;          // 4 col tiles
    int half = lane >> 4;                 // 0 or 1
    int l15  = lane & 15;

    const float* arow = x + (long)(tm + l15) * DIM + 2 * half;       // 8B aligned
    const float* bcol = w + (long)(2 * half) * NGATES + (tn + l15);

    v8f c = {};
    for (int k = 0; k < DIM; k += 4) {
        v2f a = *(const v2f*)(arow + k);           // K = k+2*half, k+2*half+1
        v2f b;
        b[0] = bcol[(long)k * NGATES];             // B[k+2*half][n]
        b[1] = bcol[(long)k * NGATES + NGATES];    // B[k+2*half+1][n]
        c = __builtin_amdgcn_wmma_f32_16x16x4_f32(false, a, false, b,
                                                  (short)0, c, false, false);
    }
    // C/D layout: VGPR r, lanes 0-15 -> M=r, lanes 16-31 -> M=r+8
#pragma unroll
    for (int r = 0; r < 8; ++r) {
        int row = tm + r + 8 * half;
        int col = tn + l15;
        logits[(long)row * NGATES + col] = c[r];
    }
}

// ---------------------------------------------------------------------------
// Kernel 3: per-token softmax + top-2 + gate normalization.
// Overwrites logits with probabilities in place (needed for density_1_proxy).
// ---------------------------------------------------------------------------
__global__ void __launch_bounds__(256) softmax_top2(float* __restrict__ probs,
                                                    int* __restrict__ idx1, int* __restrict__ idx2,
                                                    float* __restrict__ g1, float* __restrict__ g2) {
    int n = blockIdx.x * blockDim.x + threadIdx.x;
    if (n >= TOKENS) return;
    float* row = probs + (long)n * NGATES;

    float mx = row[0]; int i1 = 0;
    for (int j = 1; j < NGATES; ++j) {
        float v = row[j];
        if (v > mx) { mx = v; i1 = j; }            // strict > : first index wins (jax tie rule)
    }
    float s = 0.f, v2 = -3.4e38f; int i2 = 0;
    for (int j = 0; j < NGATES; ++j) {
        float v = row[j];
        s += __expf(v - mx);
        if (j != i1 && v > v2) { v2 = v; i2 = j; }
    }
    float inv = 1.0f / s;
    for (int j = 0; j < NGATES; ++j)
        row[j] = __expf(row[j] - mx) * inv;

    float p1 = inv;                     // exp(0)/s
    float p2 = __expf(v2 - mx) * inv;
    float den = p1 + p2 + 1e-9f;
    idx1[n] = i1; idx2[n] = i2;
    g1[n] = p1 / den; g2[n] = p2 / den;
}

// ---------------------------------------------------------------------------
// Kernel 4: per-expert stable rank over tokens (wave32 ballot scan),
// capacity masks, count outputs, and the aux-loss reduction.
// One wave per expert; positions assigned in token order.
// ---------------------------------------------------------------------------
__global__ void __launch_bounds__(32) expert_scan(const float* __restrict__ probs,
                                                  const int* __restrict__ idx1,
                                                  const int* __restrict__ idx2,
                                                  int* __restrict__ pos1, int* __restrict__ pos2,
                                                  float* __restrict__ out_tail) {
    int e    = blockIdx.x;
    int lane = threadIdx.x;
    unsigned below = (lane == 0) ? 0u : (0xFFFFFFFFu >> (32 - lane));

    int base = 0;
    for (int c = 0; c < TOKENS; c += 32) {
        int n = c + lane;
        bool m = (idx1[n] == e);
        unsigned msk = __builtin_amdgcn_ballot_w32(m);
        if (m) pos1[n] = base + __popc(msk & below);
        base += __popc(msk);
    }
    int cnt1 = base;
    int m1c  = cnt1 < CAP ? cnt1 : CAP;            // capped top-1 count

    base = m1c;                                    // top-2 positions start after capped top-1
    for (int c = 0; c < TOKENS; c += 32) {
        int n = c + lane;
        bool m = (idx2[n] == e);
        unsigned msk = __builtin_amdgcn_ballot_w32(m);
        if (m) pos2[n] = base + __popc(msk & below);
        base += __popc(msk);
    }
    int cnt2 = base - m1c;
    int room = CAP - m1c; if (room < 0) room = 0;
    int m2c  = cnt2 < room ? cnt2 : room;

    // density_1_proxy numerator: column sum of probs for this expert
    float s = 0.f;
    for (int n = lane; n < TOKENS; n += 32)
        s += probs[(long)n * NGATES + e];
    for (int off = 16; off > 0; off >>= 1)
        s += __shfl_xor(s, off, 32);

    if (lane == 0) {
        out_tail[1 + e]          = (float)m1c;     // mask_1_count
        out_tail[1 + NGATES + e] = (float)m2c;     // mask_2_count
        // loss contribution: 64 * sumP[e] * cnt1[e] / 8192^2
        float contrib = 64.0f * s * (float)cnt1 / (8192.0f * 8192.0f);
        atomicAdd(&out_tail[0], contrib);
    }
}

// ---------------------------------------------------------------------------
// Kernel 5: scatter the <=2 nonzeros per token into combine/dispatch.
// idx1 != idx2 always, so no write conflicts.
// ---------------------------------------------------------------------------
__global__ void __launch_bounds__(256) scatter(const int* __restrict__ idx1,
                                               const int* __restrict__ idx2,
                                               const int* __restrict__ pos1,
                                               const int* __restrict__ pos2,
                                               const float* __restrict__ g1,
                                               const float* __restrict__ g2,
                                               float* __restrict__ out) {
    int n = blockIdx.x * blockDim.x + threadIdx.x;
    if (n >= TOKENS) return;
    float* dispatch = out;
    float* combine  = out + (long)DISPATCH_ELEMS;

    int p1 = pos1[n];
    if (p1 < CAP) {
        long off = ((long)n * NGATES + idx1[n]) * CAP + p1;
        float v = g1[n];
        combine[off]  = v;
        dispatch[off] = (v != 0.f) ? 1.f : 0.f;
    }
    int p2 = pos2[n];
    if (p2 < CAP) {
        long off = ((long)n * NGATES + idx2[n]) * CAP + p2;
        float v = g2[n];
        combine[off]  = v;
        dispatch[off] = (v != 0.f) ? 1.f : 0.f;
    }
}

// ---------------------------------------------------------------------------
extern "C" void kernel_launch(void* const* d_in, const int* in_sizes, int n_in,
                              void* d_out, int out_size, void* d_ws, size_t ws_size,
                              hipStream_t stream) {
    const float* x = (const float*)d_in[0];   // (1, 8192, 1024) f32
    const float* w = (const float*)d_in[1];   // (1024, 64) f32
    float* out = (float*)d_out;

    char* ws = (char*)d_ws;
    float* probs = (float*)ws;                              // 8192*64 f32 = 2 MB (logits -> probs)
    int*   idx1  = (int*)(ws + (size_t)TOKENS * NGATES * 4);
    int*   idx2  = idx1 + TOKENS;
    float* g1    = (float*)(idx2 + TOKENS);
    float* g2    = g1 + TOKENS;
    int*   pos1  = (int*)(g2 + TOKENS);
    int*   pos2  = pos1 + TOKENS;

    long n4 = (long)(2 * (long)DISPATCH_ELEMS) / 4;         // 41,943,040 float4 stores
    float* tail = out + 2L * DISPATCH_ELEMS;                // [loss][m1c x64][m2c x64]

    zero_kernel  <<<2048, 256, 0, stream>>>((v4f*)out, n4, tail, 1 + 2 * NGATES);
    gemm_wmma    <<<256,  256, 0, stream>>>(x, w, probs);   // 2048 waves = 2048 tiles
    softmax_top2 <<<TOKENS / 256, 256, 0, stream>>>(probs, idx1, idx2, g1, g2);
    expert_scan  <<<NGATES, 32, 0, stream>>>(probs, idx1, idx2, pos1, pos2, tail);
    scatter      <<<TOKENS / 256, 256, 0, stream>>>(idx1, idx2, pos1, pos2, g1, g2, out);
}